// HandshakingKernel_13469017440312
// MI455X (gfx1250) — compile-verified
//
#include <hip/hip_runtime.h>
#include <math.h>

typedef __attribute__((ext_vector_type(2))) float v2f;
typedef __attribute__((ext_vector_type(4))) float v4f;
typedef __attribute__((ext_vector_type(8))) float v8f;

#define HID   768
#define SEQ   256
#define BATCH 4
#define NPAIR 32896   /* 256*257/2 */
#define MROWS 1024    /* BATCH*SEQ */

// ---------------------------------------------------------------------------
// Phase 1: LR[g] = X @ W[g*768 : (g+1)*768, :]  (g=0 -> L, g=1 -> R)
// One wave32 computes one 16x16 f32 tile via V_WMMA_F32_16X16X4_F32.
// A 16x4 layout (ISA 7.12.2): lanes 0-15 -> M=lane, K=0..1 in v[0..1];
//                             lanes 16-31 -> M=lane-16, K=2..3.
// B 4x16 mirrored; C/D: vgpr r holds M = r + 8*(lane>=16), N = lane&15.
// ---------------------------------------------------------------------------
__global__ __launch_bounds__(256) void hs_gemm_wmma(const float* __restrict__ X,
                                                    const float* __restrict__ W,
                                                    float* __restrict__ LR) {
    const int lane = threadIdx.x & 31;
    const int wave = threadIdx.x >> 5;
    const int tile = blockIdx.x * 8 + wave;      // 0 .. 6143
    const int g    = tile / (64 * 48);           // which half of W
    const int rem  = tile % (64 * 48);
    const int m0   = (rem / 48) * 16;
    const int n0   = (rem % 48) * 16;
    const int half = lane >> 4;                  // 0: K=0,1   1: K=2,3
    const int lx   = lane & 15;

    const float* __restrict__ arow = X + (size_t)(m0 + lx) * HID + 2 * half;
    const float* __restrict__ bcol = W + ((size_t)g * HID + 2 * half) * HID + n0 + lx;

    v8f c = {};
#pragma unroll 4
    for (int k = 0; k < HID; k += 4) {
        v2f a = *reinterpret_cast<const v2f*>(arow + k);   // X[m][k+2h], X[m][k+2h+1]
        v2f b;
        b.x = bcol[(size_t)(k + 0) * HID];                 // W[gh+k+2h  ][n]
        b.y = bcol[(size_t)(k + 1) * HID];                 // W[gh+k+2h+1][n]
        c = __builtin_amdgcn_wmma_f32_16x16x4_f32(false, a, false, b,
                                                  (short)0, c, false, false);
    }

    float* __restrict__ out = LR + (size_t)g * MROWS * HID;
#pragma unroll
    for (int r = 0; r < 8; ++r)
        out[(size_t)(m0 + r + 8 * half) * HID + n0 + lx] = c[r];
}

// ---------------------------------------------------------------------------
// Phase 2: out[b,p,:] = tanh(L[b,i(p),:] + R[b,j(p),:] + bias)
// L/R (6 MB) stay resident in the 192 MB L2; output stores are non-temporal
// so the 404 MB of write-once data doesn't evict them.
// ---------------------------------------------------------------------------
__device__ __forceinline__ int triu_off(int i) {     // start index of row i
    return i * SEQ - (i * (i - 1)) / 2;
}

__device__ __forceinline__ float dev_tanh(float x) {
#if __has_builtin(__builtin_amdgcn_tanhf)
    return __builtin_amdgcn_tanhf(x);                // native V_TANH_F32
#else
    return tanhf(x);
#endif
}

__global__ __launch_bounds__(192) void hs_pair_tanh(const float* __restrict__ LR,
                                                    const float* __restrict__ bias,
                                                    float* __restrict__ out) {
    const int p = blockIdx.x;
    const int b = blockIdx.y;

    // Invert upper-triangular pair index. The f32 closed form is accurate to
    // +-1 in i over p < 32896, so bounded if-fixups suffice (no scalar loops).
    int i = (int)(0.5f * (513.0f - sqrtf(513.0f * 513.0f - 8.0f * (float)p)));
    i = i < 0 ? 0 : (i > SEQ - 1 ? SEQ - 1 : i);
    if (i > 0 && triu_off(i) > p) --i;
    if (i > 0 && triu_off(i) > p) --i;
    if (i < SEQ - 1 && triu_off(i + 1) <= p) ++i;
    if (i < SEQ - 1 && triu_off(i + 1) <= p) ++i;
    const int j = i + (p - triu_off(i));

    const float* __restrict__ lrow = LR + (size_t)(b * SEQ + i) * HID;
    const float* __restrict__ rrow = LR + (size_t)MROWS * HID + (size_t)(b * SEQ + j) * HID;
    float* __restrict__ orow       = out + ((size_t)b * NPAIR + (size_t)p) * HID;

    const int d = threadIdx.x * 4;
    v4f l  = *reinterpret_cast<const v4f*>(lrow + d);
    v4f r  = *reinterpret_cast<const v4f*>(rrow + d);
    v4f bb = *reinterpret_cast<const v4f*>(bias + d);
    v4f s  = l + r + bb;
    v4f t;
    t.x = dev_tanh(s.x);
    t.y = dev_tanh(s.y);
    t.z = dev_tanh(s.z);
    t.w = dev_tanh(s.w);
    __builtin_nontemporal_store(t, reinterpret_cast<v4f*>(orow + d));
}

// ---------------------------------------------------------------------------
extern "C" void kernel_launch(void* const* d_in, const int* in_sizes, int n_in,
                              void* d_out, int out_size, void* d_ws, size_t ws_size,
                              hipStream_t stream) {
    (void)in_sizes; (void)n_in; (void)out_size; (void)ws_size;
    const float* X    = (const float*)d_in[0];   // (4,256,768)
    const float* W    = (const float*)d_in[1];   // (1536,768)
    const float* bias = (const float*)d_in[2];   // (768,)
    float*       out  = (float*)d_out;           // (4,32896,768)
    float*       LR   = (float*)d_ws;            // 2 * 1024 * 768 f32 = 6 MB

    // 6144 tiles, 8 waves (256 threads) per block
    hs_gemm_wmma<<<768, 256, 0, stream>>>(X, W, LR);

    dim3 grid2(NPAIR, BATCH);
    hs_pair_tanh<<<grid2, 192, 0, stream>>>(LR, bias, out);
}